// SNARFDeformer_39788577030670
// MI455X (gfx1250) — compile-verified
//
#include <hip/hip_runtime.h>
#include <hip/hip_bf16.h>

typedef float v2f __attribute__((ext_vector_type(2)));
typedef float v8f __attribute__((ext_vector_type(8)));

#define VSUB_N 689
#define NTILE 44            // ceil(689/16)
#define THRESH2 (0.12f * 0.12f)

// Raw v_min_num_f32: IEEE minimumNumber without the canonicalize ops LLVM
// inserts for fminf (inputs here are WMMA outputs / prior mins — fine).
__device__ __forceinline__ float min_raw(float a, float b) {
    float r;
    asm("v_min_num_f32 %0, %1, %2" : "=v"(r) : "v"(a), "v"(b));
    return r;
}

// ---------------------------------------------------------------------------
// Kernel 1: KNN (K=1) via V_WMMA_F32_16X16X4_F32.
// Each wave handles 32 points (two 16-row A-tiles sharing each B tile);
// 8 waves / 256-thread block -> 256 pts/block.
// S[m,n] = -2 p_m . v_n + |v_n|^2  (A row = [px,py,pz,1], B col = [-2v,|v|^2])
// running elementwise min over vertex tiles, lane-reduce, + |p_m|^2, compare.
// B operand is pre-swizzled in LDS into the exact WMMA-B per-lane layout:
//   lanes 0-15  : {B[K=0][n], B[K=1][n]} = {-2vx, -2vy}
//   lanes 16-31 : {B[K=2][n], B[K=3][n]} = {-2vz, |v|^2}
// B-tile loads are software-pipelined one iteration ahead (LDS table padded
// with one dummy tile) so s_wait_dscnt covers a full WMMA+min iteration.
// ---------------------------------------------------------------------------
__global__ __launch_bounds__(256) void snarf_knn_mask_kernel(
    const float* __restrict__ pts,    // [B*N, 3]
    const float* __restrict__ verts,  // [VSUB, 3]
    float* __restrict__ mask_out,     // [B*N]
    int total)
{
    __shared__ float2 sB[(NTILE + 1) * 32];   // +1 dummy tile for prefetch

    for (int t = threadIdx.x; t < (NTILE + 1) * 32; t += blockDim.x) {
        int tl = t & 31;
        int n  = ((t >> 5) << 4) + (tl & 15);
        float2 e;
        if (n < VSUB_N) {
            float vx = verts[3 * n + 0];
            float vy = verts[3 * n + 1];
            float vz = verts[3 * n + 2];
            e = (tl < 16) ? make_float2(-2.f * vx, -2.f * vy)
                          : make_float2(-2.f * vz, vx * vx + vy * vy + vz * vz);
        } else {
            // padded vertex: dot contribution 0, |v|^2 = huge -> never the min
            e = (tl < 16) ? make_float2(0.f, 0.f) : make_float2(0.f, 3.0e38f);
        }
        sB[t] = e;
    }
    __syncthreads();

    const int wave = threadIdx.x >> 5;
    const int lane = threadIdx.x & 31;
    const int base = blockIdx.x * 256 + wave * 32;   // 32 points per wave
    const int prow = lane & 15;

    const int p0 = min(base + prow,      total - 1);
    const int p1 = min(base + 16 + prow, total - 1);
    const float px0 = pts[3 * p0], py0 = pts[3 * p0 + 1], pz0 = pts[3 * p0 + 2];
    const float px1 = pts[3 * p1], py1 = pts[3 * p1 + 1], pz1 = pts[3 * p1 + 2];
    const float pn0 = px0 * px0 + py0 * py0 + pz0 * pz0;
    const float pn1 = px1 * px1 + py1 * py1 + pz1 * pz1;

    // A-matrix 16x4 f32 layout: lanes 0-15 -> K=0,1 ; lanes 16-31 -> K=2,3
    v2f a0, a1;
    a0.x = (lane < 16) ? px0 : pz0;  a0.y = (lane < 16) ? py0 : 1.0f;
    a1.x = (lane < 16) ? px1 : pz1;  a1.y = (lane < 16) ? py1 : 1.0f;

    v8f zeroc;
    v8f rmin0, rmin1;
#pragma unroll
    for (int i = 0; i < 8; ++i) { zeroc[i] = 0.0f; rmin0[i] = 3.0e38f; rmin1[i] = 3.0e38f; }

    float2 bv = sB[lane];                     // prefetch tile 0
    for (int t = 0; t < NTILE; ++t) {
        float2 bvn = sB[(t + 1) * 32 + lane]; // prefetch next tile (padded)
        v2f bm; bm.x = bv.x; bm.y = bv.y;
        v8f d0 = __builtin_amdgcn_wmma_f32_16x16x4_f32(
            false, a0, false, bm, (short)0, zeroc, false, false);
        v8f d1 = __builtin_amdgcn_wmma_f32_16x16x4_f32(
            false, a1, false, bm, (short)0, zeroc, false, false);
#pragma unroll
        for (int i = 0; i < 8; ++i) rmin0[i] = min_raw(rmin0[i], d0[i]);
#pragma unroll
        for (int i = 0; i < 8; ++i) rmin1[i] = min_raw(rmin1[i], d1[i]);
        bv = bvn;
    }

    // Reduce min across the 16 lanes of each half-wave (vertex/N dimension)
#pragma unroll
    for (int off = 1; off < 16; off <<= 1) {
#pragma unroll
        for (int i = 0; i < 8; ++i) {
            rmin0[i] = min_raw(rmin0[i], __shfl_xor(rmin0[i], off, 32));
            rmin1[i] = min_raw(rmin1[i], __shfl_xor(rmin1[i], off, 32));
        }
    }

    // rmin*[r]: lanes 0-15 hold row M=r, lanes 16-31 hold row M=r+8
#pragma unroll
    for (int r = 0; r < 8; ++r) {
        float pn0_lo = __shfl(pn0, r, 32);
        float pn0_hi = __shfl(pn0, r + 8, 32);
        float pn1_lo = __shfl(pn1, r, 32);
        float pn1_hi = __shfl(pn1, r + 8, 32);
        float d0 = rmin0[r] + ((lane < 16) ? pn0_lo : pn0_hi);
        float d1 = rmin1[r] + ((lane < 16) ? pn1_lo : pn1_hi);
        if ((lane & 15) == 0) {
            int M = r + ((lane >> 4) << 3);
            if (base + M < total)
                mask_out[base + M] = (d0 < THRESH2) ? 1.0f : 0.0f;
            if (base + 16 + M < total)
                mask_out[base + 16 + M] = (d1 < THRESH2) ? 1.0f : 0.0f;
        }
    }
}

// ---------------------------------------------------------------------------
// Kernel 2: barycentric shape offset + trilinear LBS-voxel sample +
// weight-blended transform + forward skinning. One thread per (b, n).
// lbs_voxel (25 MB) is L2-resident; tfs[b] (384 f32) staged in LDS.
// ---------------------------------------------------------------------------
__global__ __launch_bounds__(256) void snarf_deform_kernel(
    const float* __restrict__ pts_in,       // [B*N, 3]
    const float* __restrict__ shape_offset, // [V, 3]
    const int*   __restrict__ init_idx,     // [N, 3]
    const float* __restrict__ init_bar,     // [N, 3]
    const float* __restrict__ lbs_voxel,    // [24, 64, 64, 64]
    const float* __restrict__ tfs,          // [B, 24, 4, 4]
    const float* __restrict__ gscale,       // [3]
    const float* __restrict__ goffset,      // [3]
    float* __restrict__ out_pts,            // [B*N, 3]
    float* __restrict__ out_wtf,            // [B*N, 16]
    int N_, int total)
{
    __shared__ float stf[24 * 16];

    const int p = blockIdx.x * blockDim.x + threadIdx.x;
    const int b = (blockIdx.x * (int)blockDim.x) / N_;  // uniform per block

    for (int t = threadIdx.x; t < 24 * 16; t += blockDim.x)
        stf[t] = tfs[b * 24 * 16 + t];
    __syncthreads();

    if (p >= total) return;
    const int n = p - b * N_;

    // --- barycentric interpolation of per-vertex shape offsets ---
    float ox = 0.f, oy = 0.f, oz = 0.f;
#pragma unroll
    for (int k = 0; k < 3; ++k) {
        int   vi = init_idx[n * 3 + k];
        float bw = init_bar[n * 3 + k];
        ox += bw * shape_offset[vi * 3 + 0];
        oy += bw * shape_offset[vi * 3 + 1];
        oz += bw * shape_offset[vi * 3 + 2];
    }
    const float qx = pts_in[p * 3 + 0] + ox;
    const float qy = pts_in[p * 3 + 1] + oy;
    const float qz = pts_in[p * 3 + 2] + oz;

    // --- normalize to grid coords, clip to [-1,1] via v_med3_f32 ---
    float cx = __builtin_amdgcn_fmed3f((qx - goffset[0]) * gscale[0], -1.f, 1.f);
    float cy = __builtin_amdgcn_fmed3f((qy - goffset[1]) * gscale[1], -1.f, 1.f);
    float cz = __builtin_amdgcn_fmed3f((qz - goffset[2]) * gscale[2], -1.f, 1.f);

    float fx_ = (cx + 1.f) * 0.5f * 63.f;   // x -> W
    float fy_ = (cy + 1.f) * 0.5f * 63.f;   // y -> H
    float fz_ = (cz + 1.f) * 0.5f * 63.f;   // z -> D

    int x0 = (int)floorf(fx_); int x1 = min(x0 + 1, 63);
    int y0 = (int)floorf(fy_); int y1 = min(y0 + 1, 63);
    int z0 = (int)floorf(fz_); int z1 = min(z0 + 1, 63);
    float fx = fx_ - (float)x0;
    float fy = fy_ - (float)y0;
    float fz = fz_ - (float)z0;
    float gx = 1.f - fx, gy = 1.f - fy, gz = 1.f - fz;

    int o[8];
    {
        int z0y0 = z0 * 4096 + y0 * 64;
        int z0y1 = z0 * 4096 + y1 * 64;
        int z1y0 = z1 * 4096 + y0 * 64;
        int z1y1 = z1 * 4096 + y1 * 64;
        o[0] = z0y0 + x0; o[1] = z0y0 + x1;
        o[2] = z0y1 + x0; o[3] = z0y1 + x1;
        o[4] = z1y0 + x0; o[5] = z1y0 + x1;
        o[6] = z1y1 + x0; o[7] = z1y1 + x1;
    }
    float wc[8];
    wc[0] = gz * gy * gx; wc[1] = gz * gy * fx;
    wc[2] = gz * fy * gx; wc[3] = gz * fy * fx;
    wc[4] = fz * gy * gx; wc[5] = fz * gy * fx;
    wc[6] = fz * fy * gx; wc[7] = fz * fy * fx;

    // --- trilinear sample 24 channels (L2-resident gathers) ---
    float wj[24];
    float wsum = 0.f;
#pragma unroll 4
    for (int j = 0; j < 24; ++j) {
        const float* g = lbs_voxel + j * 262144;  // 64^3 per channel
        float s = wc[0] * g[o[0]] + wc[1] * g[o[1]]
                + wc[2] * g[o[2]] + wc[3] * g[o[3]]
                + wc[4] * g[o[4]] + wc[5] * g[o[5]]
                + wc[6] * g[o[6]] + wc[7] * g[o[7]];
        wj[j] = s;
        wsum += s;
    }
    const float inv = 1.0f / fmaxf(wsum, 1e-6f);

    // --- w_tf = sum_j w[j] * tfs[b, j]  (LDS broadcast reads) ---
    float acc[16];
#pragma unroll
    for (int e = 0; e < 16; ++e) acc[e] = 0.f;
    for (int j = 0; j < 24; ++j) {
        float c = wj[j] * inv;
#pragma unroll
        for (int e = 0; e < 16; ++e) acc[e] += c * stf[j * 16 + e];
    }

    // --- forward skinning: [w_tf] @ [qx,qy,qz,1] ---
    float px = acc[0] * qx + acc[1] * qy + acc[2]  * qz + acc[3];
    float py = acc[4] * qx + acc[5] * qy + acc[6]  * qz + acc[7];
    float pz = acc[8] * qx + acc[9] * qy + acc[10] * qz + acc[11];

    out_pts[p * 3 + 0] = px;
    out_pts[p * 3 + 1] = py;
    out_pts[p * 3 + 2] = pz;
#pragma unroll
    for (int e = 0; e < 16; ++e) out_wtf[p * 16 + e] = acc[e];
}

extern "C" void kernel_launch(void* const* d_in, const int* in_sizes, int n_in,
                              void* d_out, int out_size, void* d_ws, size_t ws_size,
                              hipStream_t stream) {
    const float* pts_in    = (const float*)d_in[0];  // [B,N,3]
    const float* vs_templ  = (const float*)d_in[1];  // [1,VSUB,3]
    const float* shape_off = (const float*)d_in[2];  // [1,V,3]
    const int*   init_idx  = (const int*)  d_in[3];  // [N,3]
    const float* init_bar  = (const float*)d_in[4];  // [N,3]
    const float* lbs_voxel = (const float*)d_in[5];  // [24,64,64,64]
    const float* tfs       = (const float*)d_in[6];  // [B,24,4,4]
    const float* gscale    = (const float*)d_in[7];  // [3]
    const float* goffset   = (const float*)d_in[8];  // [3]

    const int total = in_sizes[0] / 3;  // B*N = 131072
    const int N_    = total / 2;        // B = 2

    float* out      = (float*)d_out;
    float* out_pts  = out;              // [B*N, 3]
    float* out_mask = out + 3 * total;  // [B*N]
    float* out_wtf  = out + 4 * total;  // [B*N, 16]

    // Kernel 1: WMMA-based KNN -> mask
    {
        int blocks = (total + 255) / 256;  // 256 points per 256-thread block
        snarf_knn_mask_kernel<<<blocks, 256, 0, stream>>>(
            pts_in, vs_templ, out_mask, total);
    }
    // Kernel 2: offsets + trilinear LBS + blend + skinning
    {
        int blocks = (total + 255) / 256;
        snarf_deform_kernel<<<blocks, 256, 0, stream>>>(
            pts_in, shape_off, init_idx, init_bar, lbs_voxel, tfs,
            gscale, goffset, out_pts, out_wtf, N_, total);
    }
}